// MultiChannel2DCircularConv_26869315404083
// MI455X (gfx1250) — compile-verified
//
#include <hip/hip_runtime.h>

typedef float v2f __attribute__((ext_vector_type(2)));
typedef float v8f __attribute__((ext_vector_type(8)));

#define C_CH 96
#define N_SP 64
#define B_SZ 32

// ---------------------------------------------------------------------------
// Weight repack: W[c_out][c_in][ky][kx] -> Wpk[tap][c_out][c_in]
// so each lane's A-fragment K-pair (c_in, c_in+1) is one contiguous b64 load.
// ---------------------------------------------------------------------------
__global__ __launch_bounds__(256) void pack_weights(
    const float* __restrict__ W, float* __restrict__ Wpk)
{
    const int id = blockIdx.x * 256 + threadIdx.x;   // 9*96*96 = 82944 total
    if (id >= 9 * C_CH * C_CH) return;
    const int c = id % C_CH;
    const int m = (id / C_CH) % C_CH;
    const int t = id / (C_CH * C_CH);
    Wpk[id] = W[(size_t)(m * C_CH + c) * 9 + t];
}

// ---------------------------------------------------------------------------
// Conv kernel: implicit GEMM, one wave32 per 16(M) x 64(N) output strip.
//   M = c_out (96 -> 6 tiles), K = 9 taps * 96 c_in (chunks of 4),
//   N = B*64*64 (131072 -> 2048 strips of 64 = one full x-row)
// 4 accumulators share each A fragment: 4 WMMAs per A load.
// ---------------------------------------------------------------------------
__global__ __launch_bounds__(256) void conv_wmma_f32(
    const float* __restrict__ X,     // [B, C, N, N]
    const float* __restrict__ Wpk,   // [9, C, C] packed weights
    const float* __restrict__ bias,  // [1, C, N, N]
    float* __restrict__ out)         // [B, C, N, N]
{
    const int wid  = blockIdx.x * 8 + (threadIdx.x >> 5);
    const int lane = threadIdx.x & 31;
    const int l16  = lane & 15;
    const int half = lane >> 4;   // selects K-pair {0,1} vs {2,3}

    const int tile_m = wid % 6;        // group of 16 output channels
    const int strip  = wid / 6;        // 2048 strips: (batch, row)
    const int m_base = tile_m * 16;

    const int ry = strip & 63;         // output row y
    const int b  = strip >> 6;         // batch index
    const int m  = m_base + l16;       // this lane's A-matrix row

    v8f acc0 = {}, acc1 = {}, acc2 = {}, acc3 = {};

    #pragma unroll
    for (int ky = 0; ky < 3; ++ky) {
        const int yy = (ry + ky + 63) & 63;         // circular wrap
        #pragma unroll
        for (int kx = 0; kx < 3; ++kx) {
            const int t = ky * 3 + kx;
            // this lane's x source for each of the 4 sub-tiles
            const int x0 = (l16 + kx + 63) & 63;
            const int x1 = (x0 + 16) & 63;
            const int x2 = (x0 + 32) & 63;
            const int x3 = (x0 + 48) & 63;
            // input row base for c_in = 2*half
            const float* Xrow =
                X + (((size_t)(b * C_CH + 2 * half) * N_SP + yy) * N_SP);
            const float* Xp0 = Xrow + x0;
            const float* Xp1 = Xrow + x1;
            const float* Xp2 = Xrow + x2;
            const float* Xp3 = Xrow + x3;
            // packed A base: Wpk[t][m][2*half]
            const float* Wp = Wpk + ((size_t)(t * C_CH + m) * C_CH + 2 * half);

            #pragma unroll 4
            for (int c0 = 0; c0 < C_CH; c0 += 4) {
                const v2f afrag = *(const v2f*)Wp;   // one b64 load
                v2f b0, b1, b2, b3;
                b0.x = Xp0[0]; b0.y = Xp0[N_SP * N_SP];
                b1.x = Xp1[0]; b1.y = Xp1[N_SP * N_SP];
                b2.x = Xp2[0]; b2.y = Xp2[N_SP * N_SP];
                b3.x = Xp3[0]; b3.y = Xp3[N_SP * N_SP];
                acc0 = __builtin_amdgcn_wmma_f32_16x16x4_f32(
                    false, afrag, false, b0, (short)0, acc0, false, false);
                acc1 = __builtin_amdgcn_wmma_f32_16x16x4_f32(
                    false, afrag, false, b1, (short)0, acc1, false, false);
                acc2 = __builtin_amdgcn_wmma_f32_16x16x4_f32(
                    false, afrag, false, b2, (short)0, acc2, false, false);
                acc3 = __builtin_amdgcn_wmma_f32_16x16x4_f32(
                    false, afrag, false, b3, (short)0, acc3, false, false);
                Wp  += 4;                     // 4 input channels (contiguous)
                Xp0 += 4 * N_SP * N_SP;
                Xp1 += 4 * N_SP * N_SP;
                Xp2 += 4 * N_SP * N_SP;
                Xp3 += 4 * N_SP * N_SP;
            }
        }
    }

    // C/D layout: VGPR r, lanes 0-15 -> M=r ; lanes 16-31 -> M=r+8 ; N=l16
    #pragma unroll
    for (int r = 0; r < 8; ++r) {
        const int co = m_base + r + half * 8;
        const size_t obase = ((size_t)(b * C_CH + co) * N_SP + ry) * N_SP;
        const size_t bbase = ((size_t)co * N_SP + ry) * N_SP;
        out[obase + l16     ] = acc0[r] + bias[bbase + l16     ];
        out[obase + l16 + 16] = acc1[r] + bias[bbase + l16 + 16];
        out[obase + l16 + 32] = acc2[r] + bias[bbase + l16 + 32];
        out[obase + l16 + 48] = acc3[r] + bias[bbase + l16 + 48];
    }
}

// ---------------------------------------------------------------------------
// Logdet kernel: one workgroup per frequency bin (u,v).
// Builds Khat(u,v) = sum_{ky,kx} W[:,:,ky,kx] * e^{-2pi i (u*ky+v*kx)/64}
// in LDS (96x96 complex, stride 97), then partial-pivot LU accumulating
// 0.5*log(|pivot|^2). Per-bin result -> partial[bin].
// ---------------------------------------------------------------------------
__global__ __launch_bounds__(256) void logdet_lu(
    const float* __restrict__ W,   // [96, 96, 3, 3]
    float* __restrict__ partial)   // [4096]
{
    extern __shared__ float smem[];
    const int S = 97;
    float* Mre  = smem;                 // 96*97
    float* Mim  = Mre + 96 * S;         // 96*97
    float* fre  = Mim + 96 * S;         // 96
    float* fim  = fre + 96;             // 96
    float* rval = fim + 96;             // 256
    int*   ridx = (int*)(rval + 256);   // 256
    float* piv  = (float*)(ridx + 256); // [0]=invRe [1]=invIm [2]=log-acc

    const int tid = threadIdx.x;
    const int bin = blockIdx.x;
    const int u = bin >> 6;
    const int v = bin & 63;

    // twiddles for the 9 taps: e^{-i * 2pi*(u*ky+v*kx)/64}
    float ct[9], st[9];
    #pragma unroll
    for (int t = 0; t < 9; ++t) {
        const int ky = t / 3, kx = t % 3;
        const float ang = -6.2831853071795864769f * (float)(u * ky + v * kx) / 64.0f;
        __sincosf(ang, &st[t], &ct[t]);
    }

    // build the matrix
    for (int idx = tid; idx < 96 * 96; idx += 256) {
        const int i = idx / 96, j = idx % 96;
        const float* wp = W + (size_t)idx * 9;
        float re = 0.f, im = 0.f;
        #pragma unroll
        for (int t = 0; t < 9; ++t) { re += wp[t] * ct[t]; im += wp[t] * st[t]; }
        Mre[i * S + j] = re;
        Mim[i * S + j] = im;
    }
    if (tid == 0) piv[2] = 0.f;
    __syncthreads();

    for (int k = 0; k < 96; ++k) {
        // --- pivot search: argmax |M[i][k]|^2 over i in [k,96) ---
        float best = -1.f;
        int bi = k;
        const int i = k + tid;
        if (tid < 96 - k) {
            const float re = Mre[i * S + k], im = Mim[i * S + k];
            best = re * re + im * im;
            bi = i;
        }
        rval[tid] = best;
        ridx[tid] = bi;
        __syncthreads();
        for (int off = 128; off > 0; off >>= 1) {
            if (tid < off && rval[tid + off] > rval[tid]) {
                rval[tid] = rval[tid + off];
                ridx[tid] = ridx[tid + off];
            }
            __syncthreads();
        }
        const int p = ridx[0];

        // --- row swap ---
        if (p != k) {
            for (int j = k + tid; j < 96; j += 256) {
                const float tr = Mre[k * S + j];
                Mre[k * S + j] = Mre[p * S + j];
                Mre[p * S + j] = tr;
                const float ti = Mim[k * S + j];
                Mim[k * S + j] = Mim[p * S + j];
                Mim[p * S + j] = ti;
            }
        }
        __syncthreads();

        // --- pivot log + reciprocal ---
        if (tid == 0) {
            const float re = Mre[k * S + k], im = Mim[k * S + k];
            const float d = re * re + im * im;
            piv[2] += 0.5f * __logf(d);
            const float inv = 1.0f / d;
            piv[0] = re * inv;
            piv[1] = -im * inv;
        }
        __syncthreads();
        const float ivr = piv[0], ivi = piv[1];

        // --- factors f[i] = M[i][k] / pivot ---
        for (int i2 = k + 1 + tid; i2 < 96; i2 += 256) {
            const float re = Mre[i2 * S + k], im = Mim[i2 * S + k];
            fre[i2] = re * ivr - im * ivi;
            fim[i2] = re * ivi + im * ivr;
        }
        __syncthreads();

        // --- rank-1 update of trailing submatrix ---
        const int nr = 95 - k;
        const int nn = nr * nr;
        for (int t = tid; t < nn; t += 256) {
            const int i2 = k + 1 + t / nr;
            const int j2 = k + 1 + t % nr;
            const float fr = fre[i2], fi = fim[i2];
            const float br = Mre[k * S + j2], bim = Mim[k * S + j2];
            Mre[i2 * S + j2] -= fr * br - fi * bim;
            Mim[i2 * S + j2] -= fr * bim + fi * br;
        }
        __syncthreads();
    }

    if (tid == 0) partial[bin] = piv[2];
}

// ---------------------------------------------------------------------------
// Deterministic single-block reduction of the 4096 per-bin partials.
// ---------------------------------------------------------------------------
__global__ __launch_bounds__(256) void reduce_logdet(
    const float* __restrict__ partial, float* __restrict__ out,
    int n, int out_last)
{
    __shared__ float s[256];
    float acc = 0.f;
    for (int i = threadIdx.x; i < n; i += 256) acc += partial[i];
    s[threadIdx.x] = acc;
    __syncthreads();
    for (int off = 128; off > 0; off >>= 1) {
        if (threadIdx.x < off) s[threadIdx.x] += s[threadIdx.x + off];
        __syncthreads();
    }
    if (threadIdx.x == 0) out[out_last] = s[0];
}

extern "C" void kernel_launch(void* const* d_in, const int* in_sizes, int n_in,
                              void* d_out, int out_size, void* d_ws, size_t ws_size,
                              hipStream_t stream) {
    const float* X    = (const float*)d_in[0];  // conv_in  [32,96,64,64]
    const float* W    = (const float*)d_in[1];  // kernel   [96,96,3,3]
    const float* bias = (const float*)d_in[2];  // bias     [1,96,64,64]
    float* out = (float*)d_out;                 // conv_out flat + logdet scalar

    float* partial = (float*)d_ws;              // 4096 floats
    float* Wpk     = partial + 4096;            // 9*96*96 floats packed weights

    // repack weights for vectorized A-fragment loads
    pack_weights<<<(9 * C_CH * C_CH + 255) / 256, 256, 0, stream>>>(W, Wpk);

    // 12288 wave-strips (6 M-tiles x 2048 row-strips), 8 waves per block
    conv_wmma_f32<<<1536, 256, 0, stream>>>(X, Wpk, bias, out);

    const size_t lds_bytes =
        (size_t)(2 * 96 * 97 + 96 + 96 + 256) * sizeof(float) +
        256 * sizeof(int) + 3 * sizeof(float);
    logdet_lu<<<4096, 256, lds_bytes, stream>>>(W, partial);

    reduce_logdet<<<1, 256, 0, stream>>>(partial, out, 4096, out_size - 1);
}